// PyTorchCustomMSDA_26989574488776
// MI455X (gfx1250) — compile-verified
//
#include <hip/hip_runtime.h>
#include <hip/hip_bf16.h>

typedef __attribute__((ext_vector_type(2))) float v2f;
typedef __attribute__((ext_vector_type(8))) float v8f;

// ---------------------------------------------------------------------------
// Transpose [256 x N] row-major -> [N x 256] row-major (weights are tiny).
// ---------------------------------------------------------------------------
__global__ void transpose_k(const float* __restrict__ W, float* __restrict__ Wt, int N) {
  int idx = blockIdx.x * blockDim.x + threadIdx.x;
  if (idx >= 256 * N) return;
  int k = idx / N;
  int n = idx - k * N;
  Wt[(size_t)n * 256 + k] = W[idx];
}

// ---------------------------------------------------------------------------
// C[M x N] = A[M x 256] * W[256 x N] + bias[N] (+ residual[M x N])
// W passed pre-transposed: Wt[N x 256].  One wave -> one 16x16 output tile,
// K-loop of V_WMMA_F32_16X16X4_F32 (exact fp32), two independent accumulator
// chains for ILP.  A row index is CLAMPED (not masked) in the edge tile:
// out-of-range rows compute garbage that is never stored, so the hot loop is
// branch-free and loads pipeline.
// Block = 256 threads = 8 waves; wave w handles ntile = blockIdx.y*8 + w.
// ---------------------------------------------------------------------------
__global__ void gemm_wmma_k(const float* __restrict__ A, const float* __restrict__ Wt,
                            const float* __restrict__ bias, const float* __restrict__ resid,
                            float* __restrict__ C, int M, int N) {
  const int lane = threadIdx.x & 31;
  const int wv   = threadIdx.x >> 5;
  const int mt   = blockIdx.x;
  const int nt   = blockIdx.y * 8 + wv;
  const int half = lane >> 4;     // 0: k-pair {0,1}, 1: k-pair {2,3}
  const int lid  = lane & 15;
  const int m    = mt * 16 + lid; // A row for this lane's fragment
  const int n    = nt * 16 + lid; // B column for this lane's fragment
  const int mc   = (m < M) ? m : (M - 1);  // clamp: branch-free edge handling

  const float* __restrict__ arow = A  + (size_t)mc * 256 + 2 * half;
  const float* __restrict__ brow = Wt + (size_t)n  * 256 + 2 * half;

  v8f acc0 = {0.f, 0.f, 0.f, 0.f, 0.f, 0.f, 0.f, 0.f};
  v8f acc1 = {0.f, 0.f, 0.f, 0.f, 0.f, 0.f, 0.f, 0.f};

#pragma unroll 4
  for (int kk = 0; kk < 256; kk += 8) {
    v2f a0 = *(const v2f*)(arow + kk);
    v2f b0 = *(const v2f*)(brow + kk);
    v2f a1 = *(const v2f*)(arow + kk + 4);
    v2f b1 = *(const v2f*)(brow + kk + 4);
    acc0 = __builtin_amdgcn_wmma_f32_16x16x4_f32(false, a0, false, b0,
                                                 (short)0, acc0, false, false);
    acc1 = __builtin_amdgcn_wmma_f32_16x16x4_f32(false, a1, false, b1,
                                                 (short)0, acc1, false, false);
  }

  const float bn = bias[n];
#pragma unroll
  for (int i = 0; i < 8; ++i) {
    int mo = mt * 16 + i + 8 * half;   // C/D layout: row = vgpr + 8*(lane>=16)
    if (mo < M) {
      float r = resid ? resid[(size_t)mo * N + n] : 0.f;
      C[(size_t)mo * N + n] = acc0[i] + acc1[i] + bn + r;
    }
  }
}

// ---------------------------------------------------------------------------
// Per (b,q,h): softmax over 16 attn logits (in place) and convert raw offsets
// into absolute sampling locations (in place).
// off layout:  [bs*nq][h(8)][l(4)][p(4)][2]   attn layout: [bs*nq][h(8)][16]
// ---------------------------------------------------------------------------
__global__ void locattn_k(float* __restrict__ off, float* __restrict__ attn,
                          const float* __restrict__ refpts, const int* __restrict__ spatial,
                          int total) {
  int t = blockIdx.x * blockDim.x + threadIdx.x;
  if (t >= total) return;
  int bq = t >> 3;
  int h  = t & 7;
  float* __restrict__ op = off  + (size_t)bq * 256 + h * 32;
  float* __restrict__ ap = attn + (size_t)bq * 128 + h * 16;
  const float* __restrict__ rp = refpts + (size_t)bq * 8;  // [4 levels][2]

  float v[16];
  float mx = -3.0e38f;
#pragma unroll
  for (int i = 0; i < 16; ++i) { v[i] = ap[i]; mx = fmaxf(mx, v[i]); }
  float s = 0.f;
#pragma unroll
  for (int i = 0; i < 16; ++i) { v[i] = __expf(v[i] - mx); s += v[i]; }
  float inv = 1.f / s;
#pragma unroll
  for (int i = 0; i < 16; ++i) ap[i] = v[i] * inv;

#pragma unroll
  for (int l = 0; l < 4; ++l) {
    float Wf = (float)spatial[2 * l + 1];
    float Hf = (float)spatial[2 * l];
    float rx = rp[2 * l], ry = rp[2 * l + 1];
#pragma unroll
    for (int p = 0; p < 4; ++p) {
      int i = l * 4 + p;
      float ox = op[2 * i], oy = op[2 * i + 1];
      op[2 * i]     = rx + ox / Wf;
      op[2 * i + 1] = ry + oy / Hf;
    }
  }
}

// ---------------------------------------------------------------------------
// Sampling + weighted sum. One wave per (b,q,h); lane = channel d (HD=32).
// Each bilinear tap is one coalesced 128B load (32 contiguous floats of v).
// v layout: [b*nq + pixel][256] with column h*32+d.
// ---------------------------------------------------------------------------
__global__ void msda_sample_k(const float* __restrict__ v, const float* __restrict__ loc,
                              const float* __restrict__ attn, const int* __restrict__ spatial,
                              float* __restrict__ out, int nq, int total_waves) {
  int wave = (int)((blockIdx.x * blockDim.x + threadIdx.x) >> 5);
  int lane = threadIdx.x & 31;
  if (wave >= total_waves) return;
  int h  = wave & 7;
  int bq = wave >> 3;
  int b  = bq / nq;

  const float* __restrict__ lp = loc  + (size_t)bq * 256 + h * 32;
  const float* __restrict__ ap = attn + (size_t)bq * 128 + h * 16;

  float acc = 0.f;
  int rowbase = b * nq;  // start-of-level row offset into v
#pragma unroll
  for (int l = 0; l < 4; ++l) {
    int Hl = spatial[2 * l], Wl = spatial[2 * l + 1];
    float fW = (float)Wl, fH = (float)Hl;
#pragma unroll
    for (int p = 0; p < 4; ++p) {
      int i = l * 4 + p;
      float w  = ap[i];
      float x  = lp[2 * i]     * fW - 0.5f;   // ((2*loc-1)+1)*W/2 - 0.5
      float y  = lp[2 * i + 1] * fH - 0.5f;
      float x0f = floorf(x), y0f = floorf(y);
      float wx1 = x - x0f, wy1 = y - y0f;
      float wx0 = 1.f - wx1, wy0 = 1.f - wy1;
      int x0 = (int)x0f, y0 = (int)y0f;
      float tw[4] = {wx0 * wy0, wx1 * wy0, wx0 * wy1, wx1 * wy1};
      int   xs[4] = {x0, x0 + 1, x0, x0 + 1};
      int   ys[4] = {y0, y0, y0 + 1, y0 + 1};
#pragma unroll
      for (int tq = 0; tq < 4; ++tq) {
        int xi = xs[tq], yi = ys[tq];
        if (xi >= 0 && xi < Wl && yi >= 0 && yi < Hl) {
          const float* src = v + ((size_t)(rowbase + yi * Wl + xi)) * 256 + h * 32 + lane;
          acc += w * tw[tq] * src[0];
        }
      }
    }
    rowbase += Hl * Wl;
  }
  out[(size_t)bq * 256 + h * 32 + lane] = acc;
}

// ---------------------------------------------------------------------------
extern "C" void kernel_launch(void* const* d_in, const int* in_sizes, int n_in,
                              void* d_out, int out_size, void* d_ws, size_t ws_size,
                              hipStream_t stream) {
  const float* query   = (const float*)d_in[0];
  const float* value   = (const float*)d_in[1];
  const float* refpts  = (const float*)d_in[2];
  const int*   spatial = (const int*)d_in[3];
  const float* W_val   = (const float*)d_in[4];
  const float* b_val   = (const float*)d_in[5];
  const float* W_off   = (const float*)d_in[6];
  const float* b_off   = (const float*)d_in[7];
  const float* W_attn  = (const float*)d_in[8];
  const float* b_attn  = (const float*)d_in[9];
  const float* W_out   = (const float*)d_in[10];
  const float* b_out   = (const float*)d_in[11];
  float* out = (float*)d_out;

  const int bs = 2;
  const int nq = in_sizes[0] / (bs * 256);
  const int M  = bs * nq;

  float* ws      = (float*)d_ws;
  float* wt_val  = ws;                          // 256*256
  float* wt_off  = wt_val  + 65536;             // 256*256
  float* wt_attn = wt_off  + 65536;             // 128*256
  float* wt_out  = wt_attn + 32768;             // 256*256
  float* vbuf    = wt_out  + 65536;             // M*256
  float* locbuf  = vbuf    + (size_t)M * 256;   // M*256 (raw offsets -> locations, in place)
  float* attnbuf = locbuf  + (size_t)M * 256;   // M*128 (raw logits -> softmax, in place)
  float* msda    = attnbuf + (size_t)M * 128;   // M*256

  // 1) transpose weights (tiny)
  transpose_k<<<(65536 + 255) / 256, 256, 0, stream>>>(W_val,  wt_val,  256);
  transpose_k<<<(65536 + 255) / 256, 256, 0, stream>>>(W_off,  wt_off,  256);
  transpose_k<<<(32768 + 255) / 256, 256, 0, stream>>>(W_attn, wt_attn, 128);
  transpose_k<<<(65536 + 255) / 256, 256, 0, stream>>>(W_out,  wt_out,  256);

  const int mtiles = (M + 15) / 16;

  // 2) v = value @ W_val + b_val
  gemm_wmma_k<<<dim3(mtiles, 2), 256, 0, stream>>>(value, wt_val, b_val, nullptr, vbuf, M, 256);
  // 3) raw offsets = query @ W_off + b_off
  gemm_wmma_k<<<dim3(mtiles, 2), 256, 0, stream>>>(query, wt_off, b_off, nullptr, locbuf, M, 256);
  // 4) raw attn logits = query @ W_attn + b_attn
  gemm_wmma_k<<<dim3(mtiles, 1), 256, 0, stream>>>(query, wt_attn, b_attn, nullptr, attnbuf, M, 128);

  // 5) softmax + sampling locations (in place)
  int total = M * 8;
  locattn_k<<<(total + 255) / 256, 256, 0, stream>>>(locbuf, attnbuf, refpts, spatial, total);

  // 6) bilinear sampling + weighted sum -> msda
  int twaves = M * 8;
  msda_sample_k<<<((size_t)twaves * 32 + 255) / 256, 256, 0, stream>>>(
      vbuf, locbuf, attnbuf, spatial, msda, nq, twaves);

  // 7) out = msda @ W_out + b_out + query (residual fused)
  gemm_wmma_k<<<dim3(mtiles, 2), 256, 0, stream>>>(msda, wt_out, b_out, query, out, M, 256);
}